// MixedOpShared_10496900072258
// MI455X (gfx1250) — compile-verified
//
#include <hip/hip_runtime.h>

typedef __attribute__((ext_vector_type(2))) float v2f;
typedef __attribute__((ext_vector_type(8))) float v8f;

#define WAVE_SZ 32
#define CHUNK   128  // edges per wave
#define WPB     8    // waves per block (256 threads)

static __device__ inline v8f vzero8() { v8f z = {0,0,0,0,0,0,0,0}; return z; }

// LDS byte offset of a pointer into a __shared__ object (generic -> addrspace(3) -> int)
static __device__ inline unsigned lds_offset(const void* p) {
  return (unsigned)(unsigned long long)(__attribute__((address_space(3))) const char*)p;
}

// -------- kernel 1: out[i] = sum of non-positive weights (scalar broadcast term)
__global__ __launch_bounds__(256) void gnn_init_out(
    const float* __restrict__ weights, int K, float* __restrict__ out, long n)
{
  float s = 0.f;
  for (int k = 0; k < K; ++k) { float w = weights[k]; s += (w > 0.f) ? 0.f : w; }
  long i = (long)blockIdx.x * blockDim.x + threadIdx.x;
  long stride = (long)gridDim.x * blockDim.x;
  long n4 = n >> 2;
  float4* o4 = (float4*)out;
  float4 v; v.x = v.y = v.z = v.w = s;
  for (long j = i; j < n4; j += stride) o4[j] = v;
  if (i == 0) for (long j = n4 << 2; j < n; ++j) out[j] = s;
}

// -------- kernel 2: combined = sum_k max(w_k,0) * h_ops[k]  (bandwidth-bound, b128)
__global__ __launch_bounds__(256) void gnn_combine(
    const float* __restrict__ h_ops, const float* __restrict__ weights,
    int K, float* __restrict__ combined, long nd)
{
  long i = (long)blockIdx.x * blockDim.x + threadIdx.x;
  long stride = (long)gridDim.x * blockDim.x;
  if ((nd & 3) == 0) {
    long nd4 = nd >> 2;
    const float4* H = (const float4*)h_ops;
    float4* C = (float4*)combined;
    for (long j = i; j < nd4; j += stride) {
      float4 acc; acc.x = acc.y = acc.z = acc.w = 0.f;
      for (int k = 0; k < K; ++k) {
        float w = weights[k]; w = (w > 0.f) ? w : 0.f;
        float4 h = H[(long)k * nd4 + j];
        acc.x += w * h.x; acc.y += w * h.y; acc.z += w * h.z; acc.w += w * h.w;
      }
      C[j] = acc;
    }
  } else {
    for (long j = i; j < nd; j += stride) {
      float a = 0.f;
      for (int k = 0; k < K; ++k) {
        float w = weights[k]; w = (w > 0.f) ? w : 0.f;
        a += w * h_ops[(long)k * nd + j];
      }
      combined[j] = a;
    }
  }
}

// -------- kernel 3: WMMA-based sorted-COO SpMM with 16-row-slot register tiling
__global__ __launch_bounds__(256) void gnn_spmm_wmma(
    const int* __restrict__ row, const int* __restrict__ col,
    const float* __restrict__ vals, const float* __restrict__ combined,
    float* __restrict__ out, long E)
{
  __shared__ __align__(16) int   s_row[WPB][CHUNK];
  __shared__ __align__(16) int   s_col[WPB][CHUNK];
  __shared__ __align__(16) float s_val[WPB][CHUNK];
  __shared__ int s_srw[WPB][16];   // row-slot -> global row

  const int wave = threadIdx.x >> 5;
  const int lane = threadIdx.x & 31;
  const long e0 = ((long)blockIdx.x * WPB + wave) * CHUNK;

  if (e0 + CHUNK <= E) {
    // Fast path: async DMA of the three 512B edge chunks straight into LDS.
    // CHUNK*4 bytes = 32 lanes x 16B -> one b128 async op per array.
    unsigned           dR = lds_offset(&s_row[wave][0]) + lane * 16u;
    unsigned           dC = lds_offset(&s_col[wave][0]) + lane * 16u;
    unsigned           dV = lds_offset(&s_val[wave][0]) + lane * 16u;
    unsigned long long gR = (unsigned long long)(row  + e0) + lane * 16ull;
    unsigned long long gC = (unsigned long long)(col  + e0) + lane * 16ull;
    unsigned long long gV = (unsigned long long)(vals + e0) + lane * 16ull;
    asm volatile("global_load_async_to_lds_b128 %0, %1, off" :: "v"(dR), "v"(gR) : "memory");
    asm volatile("global_load_async_to_lds_b128 %0, %1, off" :: "v"(dC), "v"(gC) : "memory");
    asm volatile("global_load_async_to_lds_b128 %0, %1, off" :: "v"(dV), "v"(gV) : "memory");
    asm volatile("s_wait_asynccnt 0x0" ::: "memory");
  } else {
    // Tail chunk: manual staging with zero-value padding.
    const int lastR = row[E - 1];
    for (int t = lane; t < CHUNK; t += WAVE_SZ) {
      long e = e0 + t;
      int r, c; float v;
      if (e < E) { r = row[e]; c = col[e]; v = vals[e]; }
      else       { r = lastR;  c = 0;      v = 0.f;     }
      s_row[wave][t] = r; s_col[wave][t] = c; s_val[wave][t] = v;
    }
  }
  if (lane < 16) s_srw[wave][lane] = 0;
  __syncthreads();

  const int m = lane & 15;   // A row-slot index / B feature column
  const int h = lane >> 4;   // lane half: covers K = 2h, 2h+1

  v8f a0 = vzero8(), a1 = vzero8(), a2 = vzero8(), a3 = vzero8();
  int cur = -1, ns = 0;

  auto flushTiles = [&]() {
    __builtin_amdgcn_wave_barrier();
#pragma unroll
    for (int v = 0; v < 8; ++v) {
      int slot = v + 8 * h;                       // C layout: VGPR v -> M = v (+8 for upper half)
      long r = (long)s_srw[wave][slot];
      float* po = out + r * 64 + m;
      atomicAdd(po +  0, a0[v]);
      atomicAdd(po + 16, a1[v]);
      atomicAdd(po + 32, a2[v]);
      atomicAdd(po + 48, a3[v]);
    }
    a0 = vzero8(); a1 = vzero8(); a2 = vzero8(); a3 = vzero8();
  };

#pragma unroll 1
  for (int g = 0; g < CHUNK / 4; ++g) {
    const int base = g * 4;
    const int r0 = s_row[wave][base + 0];
    const int r1 = s_row[wave][base + 1];
    const int r2 = s_row[wave][base + 2];
    const int r3 = s_row[wave][base + 3];

    // how many new row-slots would this group open? (rows are sorted)
    int rr = cur, nc = 0;
    if (r0 != rr) { nc++; rr = r0; }
    if (r1 != rr) { nc++; rr = r1; }
    if (r2 != rr) { nc++; rr = r2; }
    if (r3 != rr) { nc++; rr = r3; }
    if (ns + nc > 16) { flushTiles(); ns = 0; cur = -1; }   // all lanes agree -> EXEC stays full

    int s0, s1, s2, s3;
    if (r0 != cur) { cur = r0; s_srw[wave][ns] = r0; s0 = ns++; } else s0 = ns - 1;
    if (r1 != cur) { cur = r1; s_srw[wave][ns] = r1; s1 = ns++; } else s1 = ns - 1;
    if (r2 != cur) { cur = r2; s_srw[wave][ns] = r2; s2 = ns++; } else s2 = ns - 1;
    if (r3 != cur) { cur = r3; s_srw[wave][ns] = r3; s3 = ns++; } else s3 = ns - 1;

    // A (16x4 f32): VGPR r, lane-half h -> K = 2h + r; row m = lane&15
    const int kx  = base + 2 * h;
    const int slx = h ? s2 : s0;
    const int sly = h ? s3 : s1;
    v2f A;
    A.x = (slx == m) ? s_val[wave][kx]     : 0.f;
    A.y = (sly == m) ? s_val[wave][kx + 1] : 0.f;

    // B (4x16 f32): VGPR r, lane-half h -> K = 2h + r; col j = lane&15
    const float* p0 = combined + (long)s_col[wave][kx]     * 64 + m;
    const float* p1 = combined + (long)s_col[wave][kx + 1] * 64 + m;
    v2f B0, B1, B2, B3;
    B0.x = p0[0];  B0.y = p1[0];
    B1.x = p0[16]; B1.y = p1[16];
    B2.x = p0[32]; B2.y = p1[32];
    B3.x = p0[48]; B3.y = p1[48];

    a0 = __builtin_amdgcn_wmma_f32_16x16x4_f32(false, A, false, B0, (short)0, a0, false, false);
    a1 = __builtin_amdgcn_wmma_f32_16x16x4_f32(false, A, false, B1, (short)0, a1, false, false);
    a2 = __builtin_amdgcn_wmma_f32_16x16x4_f32(false, A, false, B2, (short)0, a2, false, false);
    a3 = __builtin_amdgcn_wmma_f32_16x16x4_f32(false, A, false, B3, (short)0, a3, false, false);
  }
  flushTiles();
}

extern "C" void kernel_launch(void* const* d_in, const int* in_sizes, int n_in,
                              void* d_out, int out_size, void* d_ws, size_t ws_size,
                              hipStream_t stream)
{
  const float* h_ops   = (const float*)d_in[0];
  const float* weights = (const float*)d_in[1];
  const float* vals    = (const float*)d_in[2];
  const int*   rowi    = (const int*)d_in[3];
  const int*   coli    = (const int*)d_in[4];
  float* out = (float*)d_out;
  float* combined = (float*)d_ws;   // ND * 4 bytes = 25.6 MB scratch

  const int  K  = in_sizes[1];
  const long E  = in_sizes[2];
  const long ND = (long)out_size;   // N * 64

  {
    long n4 = (ND + 3) >> 2;
    int blocks = (int)((n4 + 255) / 256);
    gnn_init_out<<<blocks, 256, 0, stream>>>(weights, K, out, ND);
  }
  {
    long n4 = (ND + 3) >> 2;
    int blocks = (int)((n4 + 255) / 256);
    gnn_combine<<<blocks, 256, 0, stream>>>(h_ops, weights, K, combined, ND);
  }
  {
    long chunks = (E + CHUNK - 1) / CHUNK;
    int blocks = (int)((chunks + WPB - 1) / WPB);
    gnn_spmm_wmma<<<blocks, 256, 0, stream>>>(rowi, coli, vals, combined, out, E);
  }
}